// EGNNDynamics_20220706029884
// MI455X (gfx1250) — compile-verified
//
#include <hip/hip_runtime.h>
#include <hip/hip_bf16.h>
#include <math.h>

// ---------------- model constants ----------------
#define NLIG 24
#define NPOCK 80
#define PG 104           // nodes per graph
#define NGR 16           // graphs
#define NLT 384          // total ligand nodes
#define NTOT 1664        // total nodes
#define HID 64

typedef float v2f __attribute__((ext_vector_type(2)));
typedef float v8f __attribute__((ext_vector_type(8)));

__device__ inline v8f zero8() {
  v8f z;
  for (int i = 0; i < 8; ++i) z[i] = 0.f;
  return z;
}

// CDNA5 f32 WMMA: D[16x16] = A[16x4] * B[4x16] + C   (wave32)
__device__ inline v8f wmma4(v2f a, v2f b, v8f c) {
  return __builtin_amdgcn_wmma_f32_16x16x4_f32(false, a, false, b, (short)0, c,
                                               false, false);
}

__device__ inline float silu(float v) { return v / (1.f + __expf(-v)); }

__device__ inline int nodeIdx(int g, int k) {
  return (k < NLIG) ? g * NLIG + k : NLT + g * NPOCK + (k - NLIG);
}

// ---------------- K1: encoders + emb_in + coord copy ----------------
__global__ __launch_bounds__(64) void k_encode(
    const float* __restrict__ xh_lig, const float* __restrict__ xh_ctx,
    const float* __restrict__ t,
    const float* __restrict__ ae_w0, const float* __restrict__ ae_b0,
    const float* __restrict__ ae_w1, const float* __restrict__ ae_b1,
    const float* __restrict__ re_w0, const float* __restrict__ re_b0,
    const float* __restrict__ re_w1, const float* __restrict__ re_b1,
    const float* __restrict__ ei_w, const float* __restrict__ ei_b,
    float* __restrict__ h, float* __restrict__ x, float* __restrict__ x0) {
  int n = blockIdx.x, tid = threadIdx.x;
  bool lig = n < NLT;
  const float* row = lig ? (xh_lig + n * 19) : (xh_ctx + (n - NLT) * 24);
  int din = lig ? 16 : 21, dm = lig ? 32 : 42;
  const float* w0 = lig ? ae_w0 : re_w0;
  const float* b0 = lig ? ae_b0 : re_b0;
  const float* w1 = lig ? ae_w1 : re_w1;
  const float* b1 = lig ? ae_b1 : re_b1;
  __shared__ float f[21], mid[42], enc[16];
  if (tid < din) f[tid] = row[3 + tid];
  __syncthreads();
  if (tid < dm) {
    float s = b0[tid];
    for (int i = 0; i < din; ++i) s += f[i] * w0[i * dm + tid];
    mid[tid] = silu(s);
  }
  __syncthreads();
  if (tid < 16) {
    float s = b1[tid];
    for (int i = 0; i < dm; ++i) s += mid[i] * w1[i * 16 + tid];
    enc[tid] = s;
  }
  __syncthreads();
  int g = lig ? (n / NLIG) : ((n - NLT) / NPOCK);
  float tv = t[g];
  float s = ei_b[tid];
  for (int i = 0; i < 16; ++i) s += enc[i] * ei_w[i * 64 + tid];
  s += tv * ei_w[16 * 64 + tid];
  h[n * 64 + tid] = s;
  if (tid < 3) {
    x[n * 3 + tid] = row[tid];
    x0[n * 3 + tid] = row[tid];
  }
}

// ---------------- K2: per-node pre-GEMMs  A=h@W1[0:64], Bc=h@W1[64:128] ----
__global__ __launch_bounds__(256) void k_nodepre(const float* __restrict__ h,
                                                 const float* __restrict__ W1,
                                                 float* __restrict__ Ao,
                                                 float* __restrict__ Bo) {
  __shared__ float Wr[4096], Wc[4096];
  int tid = threadIdx.x;
  for (int i = tid; i < 4096; i += 256) {
    Wr[i] = W1[i];
    Wc[i] = W1[4096 + i];
  }
  __syncthreads();
  int wave = tid >> 5, lane = tid & 31, m = lane & 15, hh = lane >> 4;
  int tile = blockIdx.x * 8 + wave;  // 13 blocks * 8 waves = 104 tiles
  const float* hr = h + (tile * 16 + m) * 64;
  v8f accA[4], accB[4];
  for (int nt = 0; nt < 4; ++nt) {
    accA[nt] = zero8();
    accB[nt] = zero8();
  }
#pragma unroll
  for (int k = 0; k < 16; ++k) {
    int fidx = 4 * k + 2 * hh;
    v2f a;
    a.x = hr[fidx];
    a.y = hr[fidx + 1];
#pragma unroll
    for (int nt = 0; nt < 4; ++nt) {
      int col = nt * 16 + m;
      v2f br, bc;
      br.x = Wr[fidx * 64 + col];
      br.y = Wr[(fidx + 1) * 64 + col];
      bc.x = Wc[fidx * 64 + col];
      bc.y = Wc[(fidx + 1) * 64 + col];
      accA[nt] = wmma4(a, br, accA[nt]);
      accB[nt] = wmma4(a, bc, accB[nt]);
    }
  }
  for (int nt = 0; nt < 4; ++nt)
    for (int i = 0; i < 8; ++i) {
      int r = tile * 16 + i + 8 * hh;
      int fcol = nt * 16 + m;
      Ao[r * 64 + fcol] = accA[nt][i];
      Bo[r * 64 + fcol] = accB[nt][i];
    }
}

// ---------------- K3/K5: fused edge kernel --------------------------------
// mode 0: GCL edge MLP -> agg[N,64] (pre-divided by 100)
// mode 1: coord MLP    -> x_next[N,3] = x + segsum(cdiff*phi)/100
__global__ __launch_bounds__(256) void k_edge(
    const float* __restrict__ Aa, const float* __restrict__ Bc,
    const float* __restrict__ W1, const float* __restrict__ bL0,
    const float* __restrict__ W2, const float* __restrict__ bL1,
    const float* __restrict__ w3, const float* __restrict__ emb,
    const float* __restrict__ xcur, const float* __restrict__ x0,
    float* __restrict__ out, int mode) {
  __shared__ float BcS[112 * 64];
  __shared__ float AS[16 * 64];
  __shared__ float W2S[64 * 64];
  __shared__ float TS[3 * 64];
  __shared__ float wr[64], wr0[64], bS[64], w3S[64];
  __shared__ float xcS[112 * 3], x0cS[112 * 3], xrS[16 * 3], x0rS[16 * 3];

  int tid = threadIdx.x;
  int g = blockIdx.x / 7, rt = blockIdx.x % 7;

  for (int i = tid; i < 112 * 64; i += 256) {
    int c = i >> 6, o = i & 63;
    BcS[i] = (c < PG) ? Bc[nodeIdx(g, c) * 64 + o] : 0.f;
  }
  for (int i = tid; i < 16 * 64; i += 256) {
    int r = i >> 6, o = i & 63;
    int kr = rt * 16 + r;
    AS[i] = (kr < PG) ? Aa[nodeIdx(g, kr) * 64 + o] : 0.f;
  }
  for (int i = tid; i < 4096; i += 256) W2S[i] = W2[i];
  for (int i = tid; i < 112; i += 256) {
    bool v = i < PG;
    int nn = v ? nodeIdx(g, i) : 0;
    for (int d = 0; d < 3; ++d) {
      xcS[i * 3 + d] = v ? xcur[nn * 3 + d] : 0.f;
      x0cS[i * 3 + d] = v ? x0[nn * 3 + d] : 0.f;
    }
  }
  if (tid < 16) {
    int kr = rt * 16 + tid;
    bool v = kr < PG;
    int nn = v ? nodeIdx(g, kr) : 0;
    for (int d = 0; d < 3; ++d) {
      xrS[tid * 3 + d] = v ? xcur[nn * 3 + d] : 0.f;
      x0rS[tid * 3 + d] = v ? x0[nn * 3 + d] : 0.f;
    }
  }
  if (tid < 64) {
    int o = tid;
    wr[o] = W1[128 * 64 + o];
    wr0[o] = W1[129 * 64 + o];
    bS[o] = bL1[o];
    w3S[o] = (mode == 1) ? w3[o] : 0.f;
    for (int et = 0; et < 3; ++et) {
      float s = bL0[o];
      for (int jj = 0; jj < 8; ++jj) s += emb[et * 8 + jj] * W1[(130 + jj) * 64 + o];
      TS[et * 64 + o] = s;
    }
  }
  __syncthreads();

  int wave = tid >> 5, lane = tid & 31, m = lane & 15, hh = lane >> 4;
  for (int rr = 0; rr < 2; ++rr) {
    int r = wave * 2 + rr;
    int kr = rt * 16 + r;
    float sagg[4] = {0.f, 0.f, 0.f, 0.f};
    float cax = 0.f, cay = 0.f, caz = 0.f;
    float xr0 = xrS[r * 3], xr1 = xrS[r * 3 + 1], xr2 = xrS[r * 3 + 2];
    float q0 = x0rS[r * 3], q1 = x0rS[r * 3 + 1], q2 = x0rS[r * 3 + 2];
    for (int ct = 0; ct < 7; ++ct) {
      int c = ct * 16 + m;  // this lane's A-fragment column (edge col)
      float dx = xr0 - xcS[c * 3], dy = xr1 - xcS[c * 3 + 1], dz = xr2 - xcS[c * 3 + 2];
      float radial = dx * dx + dy * dy + dz * dz;
      float ex = q0 - x0cS[c * 3], ey = q1 - x0cS[c * 3 + 1], ez = q2 - x0cS[c * 3 + 2];
      float radial0 = ex * ex + ey * ey + ez * ez;
      int et = (kr < NLIG && c < NLIG) ? 1 : ((kr >= NLIG && c >= NLIG) ? 2 : 0);
      // Build SiLU(first layer) directly in WMMA A-fragment layout (registers)
      v2f ua[16];
#pragma unroll
      for (int k = 0; k < 16; ++k) {
        int fidx = 4 * k + 2 * hh;
        float p0 = AS[r * 64 + fidx] + BcS[c * 64 + fidx] + radial * wr[fidx] +
                   radial0 * wr0[fidx] + TS[et * 64 + fidx];
        float p1 = AS[r * 64 + fidx + 1] + BcS[c * 64 + fidx + 1] +
                   radial * wr[fidx + 1] + radial0 * wr0[fidx + 1] +
                   TS[et * 64 + fidx + 1];
        ua[k].x = silu(p0);
        ua[k].y = silu(p1);
      }
      float phip[8];
      if (mode == 1)
        for (int i = 0; i < 8; ++i) phip[i] = 0.f;
#pragma unroll
      for (int nt = 0; nt < 4; ++nt) {
        v8f acc = zero8();
#pragma unroll
        for (int k = 0; k < 16; ++k) {
          int fidx = 4 * k + 2 * hh;
          int col = nt * 16 + m;
          v2f bv;
          bv.x = W2S[fidx * 64 + col];
          bv.y = W2S[(fidx + 1) * 64 + col];
          acc = wmma4(ua[k], bv, acc);
        }
        float bb = bS[nt * 16 + m];
        if (mode == 0) {
          float s = 0.f;
#pragma unroll
          for (int i = 0; i < 8; ++i) {
            int cg = ct * 16 + i + 8 * hh;
            float mv = silu(acc[i] + bb);
            s += (cg < PG) ? mv : 0.f;
          }
          sagg[nt] += s;
        } else {
          float wv = w3S[nt * 16 + m];
#pragma unroll
          for (int i = 0; i < 8; ++i) phip[i] += silu(acc[i] + bb) * wv;
        }
      }
      if (mode == 1) {
#pragma unroll
        for (int i = 0; i < 8; ++i) {
          float v = phip[i];
          v += __shfl_xor(v, 1);
          v += __shfl_xor(v, 2);
          v += __shfl_xor(v, 4);
          v += __shfl_xor(v, 8);  // phi for edge-col (i+8*hh) on all 16 lanes of half
          int cg = ct * 16 + i + 8 * hh;
          if (m == 0 && cg < PG) {
            float ddx = xr0 - xcS[cg * 3], ddy = xr1 - xcS[cg * 3 + 1],
                  ddz = xr2 - xcS[cg * 3 + 2];
            float rad = ddx * ddx + ddy * ddy + ddz * ddz;
            float inv = 1.f / sqrtf(rad + 1e-8f);
            cax += ddx * inv * v;
            cay += ddy * inv * v;
            caz += ddz * inv * v;
          }
        }
      }
    }
    if (mode == 0) {
      float t0 = sagg[0] + __shfl_xor(sagg[0], 16);
      float t1 = sagg[1] + __shfl_xor(sagg[1], 16);
      float t2 = sagg[2] + __shfl_xor(sagg[2], 16);
      float t3 = sagg[3] + __shfl_xor(sagg[3], 16);
      if (hh == 0 && kr < PG) {
        int n = nodeIdx(g, kr);
        out[n * 64 + m] = t0 * 0.01f;
        out[n * 64 + 16 + m] = t1 * 0.01f;
        out[n * 64 + 32 + m] = t2 * 0.01f;
        out[n * 64 + 48 + m] = t3 * 0.01f;
      }
    } else {
      cax += __shfl_xor(cax, 16);
      cay += __shfl_xor(cay, 16);
      caz += __shfl_xor(caz, 16);
      if (lane == 0 && kr < PG) {
        int n = nodeIdx(g, kr);
        out[n * 3 + 0] = xcur[n * 3 + 0] + cax * 0.01f;
        out[n * 3 + 1] = xcur[n * 3 + 1] + cay * 0.01f;
        out[n * 3 + 2] = xcur[n * 3 + 2] + caz * 0.01f;
      }
    }
  }
}

// ---------------- K4a: u = silu(h@Wh + agg@Wa + b0) -----------------------
__global__ __launch_bounds__(256) void k_node_a(const float* __restrict__ h,
                                                const float* __restrict__ agg,
                                                const float* __restrict__ W0,
                                                const float* __restrict__ b0,
                                                float* __restrict__ U) {
  __shared__ float Wh[4096], Wa[4096];
  __shared__ float bS[64];
  int tid = threadIdx.x;
  for (int i = tid; i < 4096; i += 256) {
    Wh[i] = W0[i];
    Wa[i] = W0[4096 + i];
  }
  if (tid < 64) bS[tid] = b0[tid];
  __syncthreads();
  int wave = tid >> 5, lane = tid & 31, m = lane & 15, hh = lane >> 4;
  int tile = blockIdx.x * 8 + wave;
  const float* hr = h + (tile * 16 + m) * 64;
  const float* ar = agg + (tile * 16 + m) * 64;
  v8f acc[4];
  for (int nt = 0; nt < 4; ++nt) acc[nt] = zero8();
#pragma unroll
  for (int k = 0; k < 16; ++k) {
    int fidx = 4 * k + 2 * hh;
    v2f a1, a2;
    a1.x = hr[fidx];
    a1.y = hr[fidx + 1];
    a2.x = ar[fidx];
    a2.y = ar[fidx + 1];
#pragma unroll
    for (int nt = 0; nt < 4; ++nt) {
      int col = nt * 16 + m;
      v2f b1v, b2v;
      b1v.x = Wh[fidx * 64 + col];
      b1v.y = Wh[(fidx + 1) * 64 + col];
      b2v.x = Wa[fidx * 64 + col];
      b2v.y = Wa[(fidx + 1) * 64 + col];
      acc[nt] = wmma4(a1, b1v, acc[nt]);
      acc[nt] = wmma4(a2, b2v, acc[nt]);
    }
  }
  for (int nt = 0; nt < 4; ++nt) {
    float bb = bS[nt * 16 + m];
    for (int i = 0; i < 8; ++i) {
      int r = tile * 16 + i + 8 * hh;
      U[r * 64 + nt * 16 + m] = silu(acc[nt][i] + bb);
    }
  }
}

// ---------------- K4b: h += u@W1 + b1 -------------------------------------
__global__ __launch_bounds__(256) void k_node_b(float* __restrict__ h,
                                                const float* __restrict__ U,
                                                const float* __restrict__ W1,
                                                const float* __restrict__ b1) {
  __shared__ float Ws[4096];
  __shared__ float bS[64];
  int tid = threadIdx.x;
  for (int i = tid; i < 4096; i += 256) Ws[i] = W1[i];
  if (tid < 64) bS[tid] = b1[tid];
  __syncthreads();
  int wave = tid >> 5, lane = tid & 31, m = lane & 15, hh = lane >> 4;
  int tile = blockIdx.x * 8 + wave;
  const float* ur = U + (tile * 16 + m) * 64;
  v8f acc[4];
  for (int nt = 0; nt < 4; ++nt) acc[nt] = zero8();
#pragma unroll
  for (int k = 0; k < 16; ++k) {
    int fidx = 4 * k + 2 * hh;
    v2f a;
    a.x = ur[fidx];
    a.y = ur[fidx + 1];
#pragma unroll
    for (int nt = 0; nt < 4; ++nt) {
      int col = nt * 16 + m;
      v2f bv;
      bv.x = Ws[fidx * 64 + col];
      bv.y = Ws[(fidx + 1) * 64 + col];
      acc[nt] = wmma4(a, bv, acc[nt]);
    }
  }
  for (int nt = 0; nt < 4; ++nt) {
    float bb = bS[nt * 16 + m];
    for (int i = 0; i < 8; ++i) {
      int r = tile * 16 + i + 8 * hh;
      int fcol = nt * 16 + m;
      h[r * 64 + fcol] += acc[nt][i] + bb;
    }
  }
}

// ---------------- K7: emb_out + decoders + velocity -----------------------
__global__ __launch_bounds__(64) void k_decode(
    const float* __restrict__ h, const float* __restrict__ xf,
    const float* __restrict__ x0, const float* __restrict__ eo_w,
    const float* __restrict__ eo_b, const float* __restrict__ ad_w0,
    const float* __restrict__ ad_b0, const float* __restrict__ ad_w1,
    const float* __restrict__ ad_b1, const float* __restrict__ rd_w0,
    const float* __restrict__ rd_b0, const float* __restrict__ rd_w1,
    const float* __restrict__ rd_b1, float* __restrict__ out) {
  int n = blockIdx.x, tid = threadIdx.x;
  bool lig = n < NLT;
  __shared__ float hs[64], e[16], mid[42];
  hs[tid] = h[n * 64 + tid];
  __syncthreads();
  if (tid < 16) {
    float s = eo_b[tid];
    for (int i = 0; i < 64; ++i) s += hs[i] * eo_w[i * 17 + tid];
    e[tid] = s;
  }
  __syncthreads();
  int dm = lig ? 32 : 42, dout = lig ? 16 : 21;
  const float* w0 = lig ? ad_w0 : rd_w0;
  const float* b0 = lig ? ad_b0 : rd_b0;
  const float* w1 = lig ? ad_w1 : rd_w1;
  const float* b1 = lig ? ad_b1 : rd_b1;
  if (tid < dm) {
    float s = b0[tid];
    for (int i = 0; i < 16; ++i) s += e[i] * w0[i * dm + tid];
    mid[tid] = silu(s);
  }
  __syncthreads();
  float* orow = lig ? (out + n * 19) : (out + 7296 + (n - NLT) * 24);
  if (tid < 3) orow[tid] = xf[n * 3 + tid] - x0[n * 3 + tid];
  if (tid < dout) {
    float s = b1[tid];
    for (int i = 0; i < dm; ++i) s += mid[i] * w1[i * dout + tid];
    orow[3 + tid] = s;
  }
}

// ---------------- host-side param mapping ---------------------------------
namespace {
struct Getter {
  void* const* din;
  int idx;
  const float* big;
  long off;
  bool concat;
  const float* next(int n) {
    if (concat) {
      const float* p = big + off;
      off += n;
      return p;
    }
    (void)n;
    return (const float*)din[idx++];
  }
};

struct Weights {
  const float *ae_w0, *ae_b0, *ae_w1, *ae_b1;
  const float *ad_w0, *ad_b0, *ad_w1, *ad_b1;
  const float *re_w0, *re_b0, *re_w1, *re_b1;
  const float *rd_w0, *rd_b0, *rd_w1, *rd_b1;
  const float* emb;
  const float *ei_w, *ei_b, *eo_w, *eo_b;
  struct Blk {
    const float *e_w0[2], *e_b0[2], *e_w1[2], *e_b1[2];
    const float *n_w0[2], *n_b0[2], *n_w1[2], *n_b1[2];
    const float *c_w0, *c_b0, *c_w1, *c_b1, *c_w2;
  } blk[4];
};

static void walk_params(Getter& G, bool sorted, Weights& W) {
  auto lin = [&](const float*& w, const float*& b, int wn, int bn) {
    if (sorted) {
      b = G.next(bn);
      w = G.next(wn);
    } else {
      w = G.next(wn);
      b = G.next(bn);
    }
  };
  if (sorted) {  // jax tree_flatten: dict keys sorted alphabetically
    lin(W.ad_w0, W.ad_b0, 512, 32);
    lin(W.ad_w1, W.ad_b1, 512, 16);
    lin(W.ae_w0, W.ae_b0, 512, 32);
    lin(W.ae_w1, W.ae_b1, 512, 16);
    for (int b = 0; b < 4; ++b) {
      Weights::Blk& B = W.blk[b];
      lin(B.c_w0, B.c_b0, 8832, 64);  // 'coord' < 'gcls'
      lin(B.c_w1, B.c_b1, 4096, 64);
      B.c_w2 = G.next(64);  // final coord layer: no bias
      for (int j = 0; j < 2; ++j) {
        lin(B.e_w0[j], B.e_b0[j], 8832, 64);  // 'edge' < 'node'
        lin(B.e_w1[j], B.e_b1[j], 4096, 64);
        lin(B.n_w0[j], B.n_b0[j], 8192, 64);
        lin(B.n_w1[j], B.n_b1[j], 4096, 64);
      }
    }
    W.emb = G.next(24);
    lin(W.ei_w, W.ei_b, 1088, 64);
    lin(W.eo_w, W.eo_b, 1088, 17);
    lin(W.rd_w0, W.rd_b0, 672, 42);
    lin(W.rd_w1, W.rd_b1, 882, 21);
    lin(W.re_w0, W.re_b0, 882, 42);
    lin(W.re_w1, W.re_b1, 672, 16);
  } else {  // python dict insertion order, 'w' before 'b'
    lin(W.ae_w0, W.ae_b0, 512, 32);
    lin(W.ae_w1, W.ae_b1, 512, 16);
    lin(W.ad_w0, W.ad_b0, 512, 32);
    lin(W.ad_w1, W.ad_b1, 512, 16);
    lin(W.re_w0, W.re_b0, 882, 42);
    lin(W.re_w1, W.re_b1, 672, 16);
    lin(W.rd_w0, W.rd_b0, 672, 42);
    lin(W.rd_w1, W.rd_b1, 882, 21);
    W.emb = G.next(24);
    lin(W.ei_w, W.ei_b, 1088, 64);
    lin(W.eo_w, W.eo_b, 1088, 17);
    for (int b = 0; b < 4; ++b) {
      Weights::Blk& B = W.blk[b];
      for (int j = 0; j < 2; ++j) {
        lin(B.e_w0[j], B.e_b0[j], 8832, 64);
        lin(B.e_w1[j], B.e_b1[j], 4096, 64);
        lin(B.n_w0[j], B.n_b0[j], 8192, 64);
        lin(B.n_w1[j], B.n_b1[j], 4096, 64);
      }
      lin(B.c_w0, B.c_b0, 8832, 64);
      lin(B.c_w1, B.c_b1, 4096, 64);
      B.c_w2 = G.next(64);
    }
  }
}
}  // namespace

extern "C" void kernel_launch(void* const* d_in, const int* in_sizes, int n_in,
                              void* d_out, int out_size, void* d_ws,
                              size_t ws_size, hipStream_t stream) {
  (void)out_size;
  (void)ws_size;
  const float *xh_lig, *xh_ctx, *tin;
  Getter G;
  G.din = d_in;
  G.big = nullptr;
  G.off = 0;
  bool sortedParams = true;
  if (n_in < 20) {
    // params passed as one concatenated buffer at slot 3 (sorted-leaf order)
    xh_lig = (const float*)d_in[0];
    xh_ctx = (const float*)d_in[1];
    tin = (const float*)d_in[2];
    G.concat = true;
    G.big = (const float*)d_in[3];
    G.idx = 0;
  } else if (in_sizes[0] == 173056) {
    // fully sorted top-level: cols, edge_type, mask_context, mask_lig, params..., rows, t, xh_context, xh_lig
    tin = (const float*)d_in[110];
    xh_ctx = (const float*)d_in[111];
    xh_lig = (const float*)d_in[112];
    G.concat = false;
    G.idx = 4;
    sortedParams = (in_sizes[4] == 32);
  } else {
    // insertion order top-level: xh_lig, xh_context, t, params..., masks, rows, cols, edge_type
    xh_lig = (const float*)d_in[0];
    xh_ctx = (const float*)d_in[1];
    tin = (const float*)d_in[2];
    G.concat = false;
    G.idx = 3;
    sortedParams = (in_sizes[3] == 32);
  }
  Weights W;
  walk_params(G, sortedParams, W);

  // workspace layout (floats)
  float* ws = (float*)d_ws;
  float* h = ws;                    // 1664*64
  float* agg = h + NTOT * 64;       // 1664*64
  float* Ab = agg + NTOT * 64;      // 1664*64 (also reused as U)
  float* Bb = Ab + NTOT * 64;       // 1664*64
  float* xA = Bb + NTOT * 64;       // 1664*3
  float* xB = xA + NTOT * 3;        // 1664*3
  float* x0b = xB + NTOT * 3;       // 1664*3

  k_encode<<<NTOT, 64, 0, stream>>>(xh_lig, xh_ctx, tin, W.ae_w0, W.ae_b0,
                                    W.ae_w1, W.ae_b1, W.re_w0, W.re_b0, W.re_w1,
                                    W.re_b1, W.ei_w, W.ei_b, h, xA, x0b);

  float* xc = xA;
  float* xn = xB;
  for (int b = 0; b < 4; ++b) {
    Weights::Blk& B = W.blk[b];
    for (int j = 0; j < 2; ++j) {
      k_nodepre<<<13, 256, 0, stream>>>(h, B.e_w0[j], Ab, Bb);
      k_edge<<<112, 256, 0, stream>>>(Ab, Bb, B.e_w0[j], B.e_b0[j], B.e_w1[j],
                                      B.e_b1[j], B.e_b1[j] /*dummy w3*/, W.emb,
                                      xc, x0b, agg, 0);
      k_node_a<<<13, 256, 0, stream>>>(h, agg, B.n_w0[j], B.n_b0[j], Ab);
      k_node_b<<<13, 256, 0, stream>>>(h, Ab, B.n_w1[j], B.n_b1[j]);
    }
    k_nodepre<<<13, 256, 0, stream>>>(h, B.c_w0, Ab, Bb);
    k_edge<<<112, 256, 0, stream>>>(Ab, Bb, B.c_w0, B.c_b0, B.c_w1, B.c_b1,
                                    B.c_w2, W.emb, xc, x0b, xn, 1);
    float* tmp = xc;
    xc = xn;
    xn = tmp;
  }

  k_decode<<<NTOT, 64, 0, stream>>>(h, xc, x0b, W.eo_w, W.eo_b, W.ad_w0,
                                    W.ad_b0, W.ad_w1, W.ad_b1, W.rd_w0, W.rd_b0,
                                    W.rd_w1, W.rd_b1, (float*)d_out);
}